// ROCKET_11897059410013
// MI455X (gfx1250) — compile-verified
//
#include <hip/hip_runtime.h>

typedef float v2f __attribute__((ext_vector_type(2)));
typedef float v8f __attribute__((ext_vector_type(8)));

#define HH 224
#define WW 224
#define HWP (224*224)
#define CIN 3
#define NCHUNK 7                 // ceil(27/4) K-chunks of f32 WMMA (K=4 each)
#define NWAVE 8                  // waves per workgroup (row segments)
#define SEGROWS (HH / NWAVE)     // 28 rows per wave
#define SEGPIX  (SEGROWS * WW)   // 6272 contiguous flat pixels per segment

__device__ __forceinline__ int imax(int a, int b) { return a > b ? a : b; }
__device__ __forceinline__ int imin(int a, int b) { return a < b ? a : b; }

__global__ __launch_bounds__(NWAVE * 32) void rocket_fused(
    const float* __restrict__ X,
    const float* __restrict__ W1, const float* __restrict__ b1,
    const float* __restrict__ W2, const float* __restrict__ b2,
    const float* __restrict__ W3, const float* __restrict__ b3,
    float* __restrict__ out)
{
  __shared__ int seg[NWAVE * 16 * 4];   // per (wave, channel): cnt|pre, suf|max, sum, isum

  const int tid  = threadIdx.x;
  const int wv   = tid >> 5;       // wave = row segment
  const int lane = tid & 31;
  const int nlo  = lane & 15;
  const int hi   = lane >> 4;

  const int blk = blockIdx.x;
  const int b   = blk / 63;        // batch
  const int t   = blk % 63;        // channel tile
  const int g   = t / 21;          // group (dilation)
  const int tg  = t % 21;

  int gcount, dil, gob; const float *Wg, *bg;
  if (g == 0)      { gcount = 334; dil = 1; gob = 0;    Wg = W1; bg = b1; }
  else if (g == 1) { gcount = 333; dil = 2; gob = 1336; Wg = W2; bg = b2; }
  else             { gcount = 333; dil = 3; gob = 2668; Wg = W3; bg = b3; }

  const int  cbase  = tg * 16;
  const int  cloc   = cbase + nlo;          // this lane's channel (N dim)
  const bool cvalid = cloc < gcount;

  // --- B-matrix fragments (weights, K x N = 28 x 16; K padded 27->28) ---
  v2f wf[NCHUNK];
  #pragma unroll
  for (int kk = 0; kk < NCHUNK; ++kk) {
    #pragma unroll
    for (int j = 0; j < 2; ++j) {
      const int k = 4*kk + 2*hi + j;
      float w = 0.f;
      if (k < 27 && cvalid) w = Wg[cloc * 27 + k];   // OIHW: k = i*9 + kh*3 + kw
      wf[kk][j] = w;
    }
  }
  const float bias = cvalid ? bg[cloc] : 0.f;

  // --- per-lane patch geometry, hoisted: element offsets for the fast path,
  //     split coords for the (rare) masked border path ---
  int eoff[NCHUNK][2], ciOff[NCHUNK][2], dyA[NCHUNK][2], dxA[NCHUNK][2];
  #pragma unroll
  for (int kk = 0; kk < NCHUNK; ++kk) {
    #pragma unroll
    for (int j = 0; j < 2; ++j) {
      int k = 4*kk + 2*hi + j;
      if (k >= 27) k = 0;               // weight is zero there -> value irrelevant
      const int ci = k / 9, r = k % 9;
      const int dy = (r / 3 - 1) * dil, dx = (r % 3 - 1) * dil;
      ciOff[kk][j] = ci * HWP;
      dyA[kk][j]   = dy;
      dxA[kk][j]   = dx;
      eoff[kk][j]  = ci * HWP + dy * WW + dx;
    }
  }

  const float* Xb = X + (size_t)b * CIN * HWP;

  // per-channel segment state (authoritative in lanes 0..15 of each wave)
  int   runCur = 0, best = 0, cntT = 0, preSeg = -1, pixDone = 0;
  float sumT = 0.f, isumT = 0.f;

  const int h0 = wv * SEGROWS, h1 = h0 + SEGROWS;
  for (int h = h0; h < h1; ++h) {
    const bool yedge  = (h < dil) | (h >= HH - dil);
    const int rowbase = h * WW + nlo;
    for (int w0 = 0; w0 < WW; w0 += 16) {
      // ---- gather A fragments: batch ALL loads before any consumer ----
      v2f afr[NCHUNK];
      if (!yedge && (w0 != 0) && (w0 != WW - 16)) {
        const int base = rowbase + w0;            // interior: no masks needed
        #pragma unroll
        for (int kk = 0; kk < NCHUNK; ++kk)
          #pragma unroll
          for (int j = 0; j < 2; ++j)
            afr[kk][j] = Xb[base + eoff[kk][j]];
      } else {
        const int px = w0 + nlo;
        float raw[NCHUNK][2];
        #pragma unroll
        for (int kk = 0; kk < NCHUNK; ++kk)
          #pragma unroll
          for (int j = 0; j < 2; ++j) {
            const int y = h + dyA[kk][j], x = px + dxA[kk][j];
            raw[kk][j] = Xb[ciOff[kk][j] + imin(imax(y,0),HH-1)*WW + imin(imax(x,0),WW-1)];
          }
        #pragma unroll
        for (int kk = 0; kk < NCHUNK; ++kk)
          #pragma unroll
          for (int j = 0; j < 2; ++j) {
            const int y = h + dyA[kk][j], x = px + dxA[kk][j];
            const bool inb = (((unsigned)y < (unsigned)HH) & ((unsigned)x < (unsigned)WW));
            afr[kk][j] = inb ? raw[kk][j] : 0.f;  // zero 'same' padding
          }
      }

      // ---- conv GEMM: two independent WMMA chains to halve dependency depth ----
      v8f acc0 = {}, acc1 = {};
      #pragma unroll
      for (int kk = 0; kk < NCHUNK; ++kk) {
        if (kk & 1)
          acc1 = __builtin_amdgcn_wmma_f32_16x16x4_f32(false, afr[kk], false, wf[kk],
                                                       (short)0, acc1, false, false);
        else
          acc0 = __builtin_amdgcn_wmma_f32_16x16x4_f32(false, afr[kk], false, wf[kk],
                                                       (short)0, acc0, false, false);
      }

      // ---- features: lane owns channel nlo; pixels 8*hi + r (r=0..7), in flat order ----
      const float nb = (float)(h * WW + w0 + 8 * hi);
      int cnt8 = 0, pre8 = -1, cur = 0, mx8 = 0;
      float sum8 = 0.f, isum8 = 0.f;
      #pragma unroll
      for (int r = 0; r < 8; ++r) {
        const float v = (acc0[r] + acc1[r]) + bias;
        const bool  p = v > 0.f;
        cnt8  += p ? 1 : 0;
        sum8  += p ? v : 0.f;
        isum8 += p ? (nb + (float)r) : 0.f;
        cur = p ? (cur + 1) : 0;
        mx8 = imax(mx8, cur);
        if (!p && pre8 < 0) pre8 = r;
      }
      if (pre8 < 0) pre8 = 8;
      const int suf8 = cur;

      // stitch pixels 8..15 (held by lane nlo+16) into lane nlo's state
      const int   pk      = cnt8 | (pre8 << 8) | (suf8 << 16) | (mx8 << 24);
      const int   pk_hi   = __shfl(pk,    16 + nlo, 32);
      const float sum_hi  = __shfl(sum8,  16 + nlo, 32);
      const float isum_hi = __shfl(isum8, 16 + nlo, 32);
      const int cnt_h = pk_hi & 255, pre_h = (pk_hi >> 8) & 255,
                suf_h = (pk_hi >> 16) & 255, mx_h = (pk_hi >> 24) & 255;

      const bool all_lo = (cnt8 == 8), all_hi = (cnt_h == 8);
      const int pre16 = all_lo ? (8 + pre_h) : pre8;
      const int suf16 = all_hi ? (suf8 + 8) : suf_h;
      const int mx16  = imax(imax(mx8, mx_h), suf8 + pre_h);
      const int cnt16 = cnt8 + cnt_h;

      best   = imax(best, imax(mx16, runCur + pre16));
      if (preSeg < 0 && cnt16 != 16) preSeg = pixDone + pre16;  // segment prefix
      runCur = (cnt16 == 16) ? (runCur + 16) : suf16;
      pixDone += 16;
      cntT  += cnt16;
      sumT  += sum8 + sum_hi;
      isumT += isum8 + isum_hi;
    }
  }

  // ---- publish per-segment stats; merge segments per channel ----
  if (preSeg < 0) preSeg = pixDone;                 // whole segment positive
  const int mxS = imax(best, runCur);
  if (lane < 16) {
    const int base = (wv * 16 + nlo) * 4;
    seg[base + 0] = (cntT & 0xffff) | (preSeg << 16);
    seg[base + 1] = (runCur & 0xffff) | (mxS << 16);
    seg[base + 2] = __float_as_int(sumT);
    seg[base + 3] = __float_as_int(isumT);
  }
  __syncthreads();

  if (tid < 16 && cvalid) {                          // one lane per channel
    int runC = 0, bestC = 0, cntC = 0;
    float sumC = 0.f, isumC = 0.f;
    #pragma unroll
    for (int s = 0; s < NWAVE; ++s) {
      const int base = (s * 16 + tid) * 4;
      const int d0 = seg[base + 0], d1 = seg[base + 1];
      const float sS = __int_as_float(seg[base + 2]);
      const float iS = __int_as_float(seg[base + 3]);
      const int cntS = d0 & 0xffff, preS = (int)((unsigned)d0 >> 16);
      const int sufS = d1 & 0xffff, mxSs = (int)((unsigned)d1 >> 16);
      bestC = imax(bestC, imax(mxSs, runC + preS));
      runC  = (cntS == SEGPIX) ? (runC + SEGPIX) : sufS;
      cntC += cntS; sumC += sS; isumC += iS;
    }
    bestC = imax(bestC, runC);                       // flush trailing run

    const float cntf = (float)cntC;
    const float safe = fmaxf(cntf, 1.f);
    const float ppv  = cntf / (float)HWP;
    const float mpv  = (cntC > 0) ? (sumC  / safe) : 0.f;
    const float mipv = (cntC > 0) ? (isumC / safe) : -1.f;
    float* o = out + (size_t)b * 4000 + gob;         // [ppv|mpv|mipv|lspv] per group
    const int c = cbase + tid;
    o[0 * gcount + c] = ppv;
    o[1 * gcount + c] = mpv;
    o[2 * gcount + c] = mipv;
    o[3 * gcount + c] = (float)bestC;
  }
}

extern "C" void kernel_launch(void* const* d_in, const int* in_sizes, int n_in,
                              void* d_out, int out_size, void* d_ws, size_t ws_size,
                              hipStream_t stream) {
  const float* X  = (const float*)d_in[0];
  const float* W1 = (const float*)d_in[1];
  const float* b1 = (const float*)d_in[2];
  const float* W2 = (const float*)d_in[3];
  const float* b2 = (const float*)d_in[4];
  const float* W3 = (const float*)d_in[5];
  const float* b3 = (const float*)d_in[6];
  float* out = (float*)d_out;
  // 8 batches x 63 channel tiles; 8 waves per workgroup = row segments
  rocket_fused<<<dim3(8 * 63), dim3(NWAVE * 32), 0, stream>>>(X, W1, b1, W2, b2, W3, b3, out);
}